// ProposalLayer_27487790694778
// MI455X (gfx1250) — compile-verified
//
#include <hip/hip_runtime.h>
#include <hip/hip_bf16.h>
#include <math.h>

// ---------------- problem constants (from reference) ----------------
#define BATCH      16
#define NANCH      262144        // power of two (2^18)
#define PRE_NMS    6000
#define PROPOSALS  1000
#define CAP        8192          // sort capacity (pow2 >= PRE_NMS)
#define NMS_TH     0.7f
#define MINSZ      16.0f
#define IMGMAX     511.0f        // IMG_W-1 == IMG_H-1
#define SLOTS      6             // boxes per thread in NMS (1024*6 >= 6000)

typedef unsigned int       u32;
typedef unsigned long long u64;

// ---------------- workspace layout ----------------
constexpr size_t OFF_KEYS = 0;                                       // B*N u32      = 16 MiB
constexpr size_t OFF_META = (size_t)BATCH * NANCH * 4;               // B*4 u32 (T, need_eq, gtc, eqc)
constexpr size_t OFF_CAND = OFF_META + 4096;                         // B*CAP u64    = 1 MiB
constexpr size_t OFF_SIDX = OFF_CAND + (size_t)BATCH * CAP * 8;      // B*PRE u32
constexpr size_t OFF_SBOX = OFF_SIDX + (size_t)BATCH * PRE_NMS * 4;  // B*PRE*4 f32

// ---------------- CDNA5 Tensor Data Mover: 1D tile Global -> LDS ----------------
typedef __attribute__((ext_vector_type(4))) u32 tdm_v4u;
typedef __attribute__((ext_vector_type(8))) int tdm_v8i;
typedef __attribute__((ext_vector_type(4))) int tdm_v4i;

__device__ __forceinline__ void tdm_load_1d(u32 ldsOff, u64 gaddr, u32 dataSizeCode, u32 nElems) {
  // D# group0: count=1 | lds_addr | global_addr[56:0] | type=2 ("image")
  tdm_v4u g0;
  g0[0] = 1u;
  g0[1] = ldsOff;
  g0[2] = (u32)(gaddr & 0xFFFFFFFFull);
  g0[3] = (u32)((gaddr >> 32) & 0x01FFFFFFull) | (2u << 30);
  // D# group1: wg_mask=0, data_size, tensor_dim0=nElems, tensor_dim1=1,
  //            tile_dim0=nElems (1D), tensor_dim0_stride=nElems
  tdm_v8i g1;
  g1[0] = (int)(dataSizeCode << 16);
  g1[1] = (int)((nElems & 0xFFFFu) << 16);          // tensor_dim0[15:0] << 16
  g1[2] = (int)((1u << 16) | (nElems >> 16));       // tensor_dim1.lo=1 | tensor_dim0[31:16]
  g1[3] = (int)((nElems & 0xFFFFu) << 16);          // tile_dim0 << 16
  g1[4] = 0;                                        // tile_dim1=tile_dim2=0 (unused)
  g1[5] = (int)nElems;                              // tensor_dim0_stride[31:0]
  g1[6] = 0;
  g1[7] = 0;
  tdm_v4i g2; g2[0] = 0; g2[1] = 0; g2[2] = 0; g2[3] = 0;   // groups 2/3: dims beyond 2D unused
  tdm_v4i g3; g3[0] = 0; g3[1] = 0; g3[2] = 0; g3[3] = 0;
  tdm_v8i g4; g4[0] = 0; g4[1] = 0; g4[2] = 0; g4[3] = 0;   // extra group (6-arg toolchain form)
  g4[4] = 0; g4[5] = 0; g4[6] = 0; g4[7] = 0;
  __builtin_amdgcn_tensor_load_to_lds(g0, g1, g2, g3, g4, 0);
}

// float -> order-preserving u32 key (larger key == larger float)
__device__ __forceinline__ u32 ordkey(float f) {
  u32 u = __float_as_uint(f);
  return (u & 0x80000000u) ? ~u : (u | 0x80000000u);
}

__device__ __forceinline__ float4 rpn_box(const float4 an, const float4 dl) {
  float w  = an.z - an.x;
  float h  = an.w - an.y;
  float cx = an.x + 0.5f * w + dl.x * w;
  float cy = an.y + 0.5f * h + dl.y * h;
  float nw = w * expf(dl.z);
  float nh = h * expf(dl.w);
  float x1 = fminf(fmaxf(cx - 0.5f * nw, 0.f), IMGMAX);
  float y1 = fminf(fmaxf(cy - 0.5f * nh, 0.f), IMGMAX);
  float x2 = fminf(fmaxf(cx + 0.5f * nw, 0.f), IMGMAX);
  float y2 = fminf(fmaxf(cy + 0.5f * nh, 0.f), IMGMAX);
  return make_float4(x1, y1, x2, y2);
}

// ---------------- K1: decode boxes, compute masked-score keys (streaming) ----------------
__global__ void proposal_key_kernel(const float* __restrict__ anchors,
                                    const float* __restrict__ probs,
                                    const float* __restrict__ deltas,
                                    u32* __restrict__ keys) {
  size_t t = (size_t)blockIdx.x * blockDim.x + threadIdx.x;   // over BATCH*NANCH
  int a = (int)(t & (size_t)(NANCH - 1));
  float4 an = reinterpret_cast<const float4*>(anchors)[a];
  float4 dl = reinterpret_cast<const float4*>(deltas)[t];
  __builtin_prefetch(&deltas[4 * t + 4096], 0, 0);            // global_prefetch_b8
  float sc = probs[2 * t + 1];
  float4 bx = rpn_box(an, dl);
  bool valid = (bx.z - bx.x >= MINSZ) && (bx.w - bx.y >= MINSZ);
  float m = valid ? sc : -__builtin_inff();
  keys[t] = ordkey(m);
}

// ---------------- K2: per-image MSB radix select of the 6000th-largest key ----------------
__global__ __launch_bounds__(1024) void proposal_select_kernel(const u32* __restrict__ keys,
                                                               u32* __restrict__ meta) {
  __shared__ u32 hist[256];
  __shared__ u32 sh_prefix, sh_k;
  int b = blockIdx.x;
  const u32* kimg = keys + (size_t)b * NANCH;
  if (threadIdx.x == 0) { sh_prefix = 0u; sh_k = PRE_NMS; }
  for (int p = 3; p >= 0; --p) {
    for (int i = threadIdx.x; i < 256; i += blockDim.x) hist[i] = 0u;
    __syncthreads();
    u32 prefix = sh_prefix;
    u32 pmask  = (p == 3) ? 0u : (0xFFFFFFFFu << ((p + 1) * 8));
    for (int i = threadIdx.x; i < NANCH; i += blockDim.x) {
      u32 key = kimg[i];
      if ((key & pmask) == prefix) atomicAdd(&hist[(key >> (p * 8)) & 0xFFu], 1u);
    }
    __syncthreads();
    if (threadIdx.x == 0) {
      u32 k = sh_k, cum = 0u; int chosen = 0;
      for (int bkt = 255; bkt >= 0; --bkt) {
        cum += hist[bkt];
        if (cum >= k) { chosen = bkt; sh_k = k - (cum - hist[bkt]); break; }
      }
      sh_prefix = prefix | ((u32)chosen << (p * 8));
    }
    __syncthreads();
  }
  if (threadIdx.x == 0) {
    meta[b * 4 + 0] = sh_prefix;  // threshold key T
    meta[b * 4 + 1] = sh_k;       // need_eq: # keys == T still wanted
    meta[b * 4 + 2] = 0u;         // gtc
    meta[b * 4 + 3] = 0u;         // eqc
  }
}

// ---------------- K3: compact candidates (key,~idx) packed u64 ----------------
__global__ void proposal_compact_kernel(const u32* __restrict__ keys,
                                        u32* __restrict__ meta,
                                        u64* __restrict__ cand) {
  size_t t = (size_t)blockIdx.x * blockDim.x + threadIdx.x;
  int b = (int)(t >> 18);                    // / NANCH
  u32 a = (u32)(t & (size_t)(NANCH - 1));
  u32 key = keys[t];
  u32 T = meta[b * 4 + 0];
  u32 need_eq = meta[b * 4 + 1];
  u32 count_gt = PRE_NMS - need_eq;
  u64 pack = ((u64)key << 32) | (u64)(~a);   // key desc, idx asc after u64-desc sort
  if (key > T) {
    u32 pos = atomicAdd(&meta[b * 4 + 2], 1u);
    cand[(size_t)b * CAP + pos] = pack;
  } else if (key == T) {
    u32 e = atomicAdd(&meta[b * 4 + 3], 1u);
    if (e < (u32)CAP - count_gt) cand[(size_t)b * CAP + count_gt + e] = pack;
  }
}

// ---------------- K4: per-image bitonic sort (desc) of 8192 packed candidates ----------------
__global__ __launch_bounds__(1024) void proposal_sort_kernel(const u64* __restrict__ cand,
                                                             const u32* __restrict__ meta,
                                                             u32* __restrict__ sidx) {
  __shared__ u64 s[CAP];                     // 64 KB LDS
  int b = blockIdx.x;
  u32 need_eq = meta[b * 4 + 1];
  u32 count_gt = PRE_NMS - need_eq;
  u32 eq = meta[b * 4 + 3];
  u32 cap_eq = (u32)CAP - count_gt;
  u32 filled = count_gt + (eq < cap_eq ? eq : cap_eq);
  // TDM: one DMA pulls the whole candidate block into LDS (ASYNC via TENSORcnt)
  if (threadIdx.x < 32u) {
    tdm_load_1d((u32)(uintptr_t)(void*)&s[0],
                (u64)(uintptr_t)(const void*)(cand + (size_t)b * CAP),
                3u /*8B*/, (u32)CAP);
    __builtin_amdgcn_s_wait_tensorcnt(0);
  }
  __syncthreads();
  for (int i = threadIdx.x; i < CAP; i += blockDim.x)
    if ((u32)i >= filled) s[i] = 0ull;       // pads sort last
  __syncthreads();
  for (int k = 2; k <= CAP; k <<= 1) {
    for (int j = k >> 1; j > 0; j >>= 1) {
      for (int i = threadIdx.x; i < CAP; i += blockDim.x) {
        int ixj = i ^ j;
        if (ixj > i) {
          u64 A = s[i], Bv = s[ixj];
          bool firstHalf = ((i & k) == 0);
          if (firstHalf ? (A < Bv) : (A > Bv)) { s[i] = Bv; s[ixj] = A; }
        }
      }
      __syncthreads();
    }
  }
  for (int i = threadIdx.x; i < PRE_NMS; i += blockDim.x)
    sidx[(size_t)b * PRE_NMS + i] = ~((u32)(s[i] & 0xFFFFFFFFull));
}

// ---------------- K5: gather sorted boxes (recompute decode for 6000/image) ----------------
__global__ void proposal_gather_kernel(const float* __restrict__ anchors,
                                       const float* __restrict__ deltas,
                                       const u32* __restrict__ sidx,
                                       float* __restrict__ sboxes) {
  int t = blockIdx.x * blockDim.x + threadIdx.x;   // BATCH*PRE_NMS == 96000 exactly
  int b = t / PRE_NMS;
  u32 a = sidx[t];
  float4 an = reinterpret_cast<const float4*>(anchors)[a];
  float4 dl = reinterpret_cast<const float4*>(deltas)[(size_t)b * NANCH + a];
  reinterpret_cast<float4*>(sboxes)[t] = rpn_box(an, dl);
}

// ---------------- K6: NMS + compaction. Boxes TDM'd into LDS; per-lane state in VGPRs ----------------
__global__ __launch_bounds__(1024) void proposal_nms_kernel(const float* __restrict__ sboxes,
                                                            float* __restrict__ out) {
  extern __shared__ char smem[];
  float* bx = (float*)smem;                              // PRE_NMS*4 floats = 96000 B
  u32* flagL = (u32*)(smem + PRE_NMS * 16);              // pivot-active broadcast
  u32* scanL = (u32*)(smem + PRE_NMS * 16 + 64);         // 1024 u32 block scan
  int b = blockIdx.x, t = threadIdx.x;

  for (int i = t; i < PROPOSALS * 4; i += 1024)          // zero-pad output
    out[(size_t)b * PROPOSALS * 4 + i] = 0.f;

  if (t < 32) {                                          // one TDM DMA: 96 KB into 320 KB WGP LDS
    tdm_load_1d((u32)(uintptr_t)(void*)bx,
                (u64)(uintptr_t)(const void*)(sboxes + (size_t)b * PRE_NMS * 4),
                2u /*4B*/, (u32)(PRE_NMS * 4));
    __builtin_amdgcn_s_wait_tensorcnt(0);
  }
  __syncthreads();

  float x1[SLOTS], y1[SLOTS], x2[SLOTS], y2[SLOTS], ar[SLOTS];
  bool sup[SLOTS], keep[SLOTS];
#pragma unroll
  for (int kk = 0; kk < SLOTS; ++kk) {
    int s = t * SLOTS + kk;
    keep[kk] = false;
    if (s < PRE_NMS) {
      x1[kk] = bx[4 * s + 0]; y1[kk] = bx[4 * s + 1];
      x2[kk] = bx[4 * s + 2]; y2[kk] = bx[4 * s + 3];
      float w = x2[kk] - x1[kk], h = y2[kk] - y1[kk];
      ar[kk] = w * h;
      sup[kk] = !(w >= MINSZ && h >= MINSZ);             // invalid start suppressed
    } else {
      x1[kk] = y1[kk] = x2[kk] = y2[kk] = ar[kk] = 0.f;
      sup[kk] = true;
    }
  }

  for (int i = 0; i < PRE_NMS; ++i) {
    int ot = i / SLOTS;
    if (t == ot) {
      int ok = i - ot * SLOTS;
      u32 act = 0u;
#pragma unroll
      for (int kk = 0; kk < SLOTS; ++kk)
        if (kk == ok) { act = sup[kk] ? 0u : 1u; if (act) keep[kk] = true; }
      flagL[0] = act;
    }
    __syncthreads();
    if (flagL[0]) {
      float px1 = bx[4 * i + 0], py1 = bx[4 * i + 1];    // uniform LDS broadcast reads
      float px2 = bx[4 * i + 2], py2 = bx[4 * i + 3];
      float pa = (px2 - px1) * (py2 - py1);
#pragma unroll
      for (int kk = 0; kk < SLOTS; ++kk) {
        int s = t * SLOTS + kk;
        if (s > i && s < PRE_NMS && !sup[kk]) {
          float xx1 = fmaxf(px1, x1[kk]);
          float yy1 = fmaxf(py1, y1[kk]);
          float xx2 = fminf(px2, x2[kk]);
          float yy2 = fminf(py2, y2[kk]);
          float inter = fmaxf(xx2 - xx1, 0.f) * fmaxf(yy2 - yy1, 0.f);
          float iou = inter / (pa + ar[kk] - inter + 1e-9f);
          if (iou > NMS_TH) sup[kk] = true;
        }
      }
    }
    __syncthreads();
  }

  // ordered compaction: per-thread count -> block scan -> write first 1000 kept
  u32 cnt = 0;
#pragma unroll
  for (int kk = 0; kk < SLOTS; ++kk)
    if (t * SLOTS + kk < PRE_NMS && keep[kk]) cnt++;
  scanL[t] = cnt;
  __syncthreads();
  for (int off = 1; off < 1024; off <<= 1) {
    u32 v = (t >= off) ? scanL[t - off] : 0u;
    __syncthreads();
    scanL[t] += v;
    __syncthreads();
  }
  u32 r = scanL[t] - cnt;                                // exclusive prefix (slot order == thread order)
#pragma unroll
  for (int kk = 0; kk < SLOTS; ++kk) {
    int s = t * SLOTS + kk;
    if (s < PRE_NMS && keep[kk]) {
      if (r < PROPOSALS)
        reinterpret_cast<float4*>(out + (size_t)b * PROPOSALS * 4)[r] =
            make_float4(x1[kk], y1[kk], x2[kk], y2[kk]);
      r++;
    }
  }
}

// ---------------- host ----------------
extern "C" void kernel_launch(void* const* d_in, const int* in_sizes, int n_in,
                              void* d_out, int out_size, void* d_ws, size_t ws_size,
                              hipStream_t stream) {
  (void)in_sizes; (void)n_in; (void)out_size; (void)ws_size;
  const float* anchors = (const float*)d_in[0];   // (NANCH,4)
  const float* probs   = (const float*)d_in[1];   // (BATCH,NANCH,2)
  const float* deltas  = (const float*)d_in[2];   // (BATCH,NANCH,4)
  float* out = (float*)d_out;                     // (BATCH,PROPOSALS,4)

  char* ws = (char*)d_ws;
  u32*  keys   = (u32*)(ws + OFF_KEYS);
  u32*  meta   = (u32*)(ws + OFF_META);
  u64*  cand   = (u64*)(ws + OFF_CAND);
  u32*  sidx   = (u32*)(ws + OFF_SIDX);
  float* sboxes = (float*)(ws + OFF_SBOX);

  const int total = BATCH * NANCH;
  proposal_key_kernel<<<total / 256, 256, 0, stream>>>(anchors, probs, deltas, keys);
  proposal_select_kernel<<<BATCH, 1024, 0, stream>>>(keys, meta);
  proposal_compact_kernel<<<total / 256, 256, 0, stream>>>(keys, meta, cand);
  proposal_sort_kernel<<<BATCH, 1024, 0, stream>>>(cand, meta, sidx);
  proposal_gather_kernel<<<(BATCH * PRE_NMS) / 256, 256, 0, stream>>>(anchors, deltas, sidx, sboxes);
  size_t nmsSmem = (size_t)PRE_NMS * 16 + 64 + 1024 * 4;  // boxes + flag + scan (~100 KB of 320 KB WGP LDS)
  proposal_nms_kernel<<<BATCH, 1024, nmsSmem, stream>>>(sboxes, out);
}